// hmcf_9311489098325
// MI455X (gfx1250) — compile-verified
//
#include <hip/hip_runtime.h>
#include <hip/hip_bf16.h>
#include <math.h>

typedef __attribute__((ext_vector_type(2))) float v2f;
typedef __attribute__((ext_vector_type(8))) float v8f;

#define D 64
#define OUTW 129  // 64 (out) + 1 (cosh) + 64 (sinh part)

// ---------------------------------------------------------------------------
// K1: x_a = concat(user, item); x_b = 0; out[:,0:64] = x (layer-0 term); deg = 0
// ---------------------------------------------------------------------------
__global__ __launch_bounds__(256) void init_kernel(
    const float* __restrict__ user_emb, const float* __restrict__ item_emb,
    float* __restrict__ x, float* __restrict__ xz, unsigned* __restrict__ deg,
    float* __restrict__ out, int n_users, int N)
{
    size_t idx = (size_t)blockIdx.x * blockDim.x + threadIdx.x;
    size_t total = (size_t)N * D;
    if (idx < total) {
        size_t n = idx >> 6;
        int    c = (int)(idx & 63);
        float v = (n < (size_t)n_users) ? user_emb[idx]
                                        : item_emb[idx - (size_t)n_users * D];
        x[idx]  = v;
        xz[idx] = 0.0f;
        out[n * OUTW + c] = v;
    }
    if (idx < (size_t)N) deg[idx] = 0u;
}

// ---------------------------------------------------------------------------
// K2: degree histogram (exact, u32 atomics)
// ---------------------------------------------------------------------------
__global__ __launch_bounds__(256) void degree_kernel(
    const int* __restrict__ h, unsigned* __restrict__ deg, int E)
{
    int e = blockIdx.x * blockDim.x + threadIdx.x;
    if (e < E) atomicAdd(&deg[h[e]], 1u);
}

// ---------------------------------------------------------------------------
// K3: one propagation layer: xdst[h[e],:] += rsqrt(dh*dt) * xsrc[t[e],:]
// 16 threads per edge, 4 columns each (float4 gather + 4 f32 atomics).
// x (64 MB) is L2-resident on MI455X (192 MB L2), so gathers + atomics stay
// on-chip.
// ---------------------------------------------------------------------------
__global__ __launch_bounds__(256) void scatter_kernel(
    const int* __restrict__ h, const int* __restrict__ t,
    const unsigned* __restrict__ deg,
    const float* __restrict__ xsrc, float* __restrict__ xdst, int E)
{
    size_t gid = (size_t)blockIdx.x * blockDim.x + threadIdx.x;
    size_t e = gid >> 4;
    if (e >= (size_t)E) return;
    int c = (int)(gid & 15) << 2;

    int hv = h[e];
    int tv = t[e];
    unsigned dh = deg[hv]; if (dh < 1u) dh = 1u;
    unsigned dt = deg[tv]; if (dt < 1u) dt = 1u;
    float g = rsqrtf((float)dh * (float)dt);

    const float4 v = *(const float4*)(xsrc + (size_t)tv * D + c);
    float* dp = xdst + (size_t)hv * D + c;
    atomicAdd(dp + 0, g * v.x);
    atomicAdd(dp + 1, g * v.y);
    atomicAdd(dp + 2, g * v.z);
    atomicAdd(dp + 3, g * v.w);
}

// ---------------------------------------------------------------------------
// K4: out[:,0:64] += xnew; zero the other ping-pong buffer for the next layer
// ---------------------------------------------------------------------------
__global__ __launch_bounds__(256) void accum_kernel(
    const float* __restrict__ xnew, float* __restrict__ xzero,
    float* __restrict__ out, int N)
{
    size_t idx = (size_t)blockIdx.x * blockDim.x + threadIdx.x;
    if (idx < (size_t)N * D) {
        size_t n = idx >> 6;
        int    c = (int)(idx & 63);
        out[n * OUTW + c] += xnew[idx];
        xzero[idx] = 0.0f;
    }
}

// ---------------------------------------------------------------------------
// K5: epilogue. Per wave: 16-row tile, sumsq via WMMA diag(X * X^T) in exact
// f32 (V_WMMA_F32_16X16X4_F32, 16 chained MACs over d=64).
// Layout trick: for this instruction the 16x4 f32 A-tile (lane=M, half->K)
// and the 4x16 f32 B-tile (lane=N, half->K) have identical register images
// when B = A^T, so the same v2f operand serves both A and B.
// Then theta = sqrt(sumsq)/3 (clamped), and phase B writes
// [out/3, cosh(theta), sinh(theta)/theta * out/3].
// ---------------------------------------------------------------------------
__global__ __launch_bounds__(256) void finalize_kernel(float* __restrict__ out, int N)
{
    __shared__ float s_theta[128];
    const int tid  = threadIdx.x;
    const int lane = tid & 31;
    const int wave = tid >> 5;
    const long blockRow0 = (long)blockIdx.x * 128;
    const long tile0     = blockRow0 + (long)wave * 16;

    if (tile0 < (long)N) {               // wave-uniform: EXEC stays all-ones
        const int r     = lane & 15;     // row within tile (M and N index)
        const int khalf = lane >> 4;     // lane-half selects K pair
        long row = tile0 + r;
        if (row >= (long)N) row = (long)N - 1;   // unused when N%16==0
        const float* rp = out + row * OUTW;

        v8f acc = {0.f, 0.f, 0.f, 0.f, 0.f, 0.f, 0.f, 0.f};
#pragma unroll
        for (int k0 = 0; k0 < D; k0 += 4) {
            v2f a;
            a.x = rp[k0 + 2 * khalf + 0];
            a.y = rp[k0 + 2 * khalf + 1];
            // D += A * A^T  (same registers for A and B by layout identity)
            acc = __builtin_amdgcn_wmma_f32_16x16x4_f32(
                false, a, false, a, (short)0, acc, false, false);
        }

        // Extract diagonal: D[m][m] lives at (VGPR m, lane m) for m<8 and
        // (VGPR m-8, lane m+16) for m>=8. Broadcast to lane m (m = 0..15).
        float sumsq = 0.0f;
#pragma unroll
        for (int v = 0; v < 8; ++v) {
            float lo = __shfl(acc[v], v, 32);
            float hi = __shfl(acc[v], v + 24, 32);
            if (lane == v)     sumsq = lo;
            if (lane == v + 8) sumsq = hi;
        }
        if (lane < 16) {
            float theta = sqrtf(sumsq) * (1.0f / 3.0f);   // ||raw||/3 = ||out||
            theta = fmaxf(theta, 1e-7f);
            s_theta[wave * 16 + lane] = theta;
        }
    }
    __syncthreads();

    // Phase B: rewrite rows in place.
    for (int idx = tid; idx < 128 * D; idx += 256) {
        int  rl = idx >> 6;
        int  cc = idx & 63;
        long n  = blockRow0 + rl;
        if (n >= (long)N) continue;
        float theta = s_theta[rl];
        float* rp = out + n * OUTW;
        float raw = rp[cc];
        float ov  = raw * (1.0f / 3.0f);
        float sh  = sinhf(theta);
        rp[cc]       = ov;
        rp[65 + cc]  = (sh / theta) * ov;
        if (cc == 0) rp[64] = coshf(theta);
    }
}

// ---------------------------------------------------------------------------
extern "C" void kernel_launch(void* const* d_in, const int* in_sizes, int n_in,
                              void* d_out, int out_size, void* d_ws, size_t ws_size,
                              hipStream_t stream)
{
    const float* user_emb = (const float*)d_in[0];
    const float* item_emb = (const float*)d_in[1];
    const int*   h_list   = (const int*)d_in[2];
    const int*   t_list   = (const int*)d_in[3];
    float* out = (float*)d_out;

    const int n_users = in_sizes[0] / D;
    const int n_items = in_sizes[1] / D;
    const int N = n_users + n_items;
    const int E = in_sizes[2];

    // scratch: x_a [N*64 f32] | x_b [N*64 f32] | deg [N u32]  (~129 MB)
    float*    x_a = (float*)d_ws;
    float*    x_b = x_a + (size_t)N * D;
    unsigned* deg = (unsigned*)(x_b + (size_t)N * D);

    const int B = 256;
    const size_t totNd  = (size_t)N * D;
    const size_t totE16 = (size_t)E * 16;
    const unsigned gNd  = (unsigned)((totNd  + B - 1) / B);
    const unsigned gE   = (unsigned)(((size_t)E + B - 1) / B);
    const unsigned gE16 = (unsigned)((totE16 + B - 1) / B);

    init_kernel<<<gNd, B, 0, stream>>>(user_emb, item_emb, x_a, x_b, deg, out, n_users, N);
    degree_kernel<<<gE, B, 0, stream>>>(h_list, deg, E);

    // layer 1: x_a -> x_b ; out += x_b ; zero x_a
    scatter_kernel<<<gE16, B, 0, stream>>>(h_list, t_list, deg, x_a, x_b, E);
    accum_kernel<<<gNd, B, 0, stream>>>(x_b, x_a, out, N);

    // layer 2: x_b -> x_a ; out += x_a ; zero x_b
    scatter_kernel<<<gE16, B, 0, stream>>>(h_list, t_list, deg, x_b, x_a, E);
    accum_kernel<<<gNd, B, 0, stream>>>(x_a, x_b, out, N);

    // epilogue: /3 + hyperboloid lift, sumsq via f32 WMMA
    finalize_kernel<<<(unsigned)((N + 127) / 128), 256, 0, stream>>>(out, N);
}